// NSM_87153476370804
// MI455X (gfx1250) — compile-verified
//
#include <hip/hip_runtime.h>

// ---------------------------------------------------------------------------
// Reference: Wx = x@W^T ; 100x ( y = Wx - y@M ).
// Linear recurrence with constant M  =>  factor x out:
//   Z_0 = W^T (1024x512);  Z_{t+1} = W^T + Z_t @ (-M);  y = x @ Z_100.
// K1: pack -M so each lane's B pair is one b64.
// K2: 100 fused iterations on Z (rows independent, 16-row blocks in LDS,
//     double buffered); writes Z_100 in pair-packed B layout.
// K3: y = x @ Z_100  -- pure WMMA GEMM, 16 acc tiles/wave, no LDS.
// All matrix math: V_WMMA_F32_16X16X4_F32 (CDNA5 wave32).
// ---------------------------------------------------------------------------

typedef float v2f __attribute__((ext_vector_type(2)));
typedef float v8f __attribute__((ext_vector_type(8)));

#define BATCH    16384
#define IN_DIM   1024
#define OUT_DIM  512
#define ITERS    100
#define ZSTRIDE  516      // 512 + 4 pad floats -> conflict-free LDS

__device__ __forceinline__ v8f wmma4(v2f a, v2f b, v8f c) {
    return __builtin_amdgcn_wmma_f32_16x16x4_f32(
        false, a, false, b, (short)0, c, false, false);
}

// K1: mneg[((r>>1)*512 + n)*2 + (r&1)] = -M[r*512 + n]
__global__ __launch_bounds__(256)
void NSM_packM(const float* __restrict__ M, float* __restrict__ mneg) {
    const int idx = blockIdx.x * 256 + threadIdx.x;   // 0 .. 512*512-1
    const int r = idx >> 9, n = idx & 511;
    mneg[((((r >> 1) << 9) | n) << 1) | (r & 1)] = -M[idx];
}

// K2: Z-iteration. 64 WGs x 16 rows; 8 waves each own 4 col-blocks.
__global__ __launch_bounds__(256, 1)
void NSM_zIter(const float* __restrict__ W,
               const float* __restrict__ mneg,
               float* __restrict__ zpack)
{
    extern __shared__ float lds[];
    float* buf0 = lds;                 // Z (even iterations)
    float* buf1 = lds + 16 * ZSTRIDE;  // Z (odd iterations)

    const int tid  = threadIdx.x;
    const int wave = tid >> 5;
    const int lane = tid & 31;
    const int half = lane >> 4;
    const int lq   = lane & 15;
    const int row0 = blockIdx.x * 16;  // Z row block (0..1023)

    // Z_0 tiles = W^T: wt[c][j] = W[n0c+lq][row0 + j + 8*half]  (v8f load)
    v8f wt[4], acc[4];
    #pragma unroll
    for (int c = 0; c < 4; ++c) {
        const int colg = (wave * 4 + c) * 16 + lq;       // Z column
        wt[c] = *(const v8f*)(W + (size_t)colg * IN_DIM + row0 + 8 * half);
        #pragma unroll
        for (int j = 0; j < 8; ++j)
            buf0[(j + 8 * half) * ZSTRIDE + colg] = wt[c][j];
    }
    __syncthreads();

    for (int it = 0; it < ITERS; ++it) {
        const float* src = (it & 1) ? buf1 : buf0;
        float*       dst = (it & 1) ? buf0 : buf1;

        #pragma unroll
        for (int c = 0; c < 4; ++c) acc[c] = wt[c];      // C init = W^T tile

        const float* sp = src + lq * ZSTRIDE + 2 * half;
        const float* mp = mneg + ((size_t)half * OUT_DIM + wave * 64 + lq) * 2;

        #pragma unroll 4
        for (int k = 0; k < OUT_DIM; k += 4) {
            v2f a = *(const v2f*)(sp + k);
            v2f b[4];
            #pragma unroll
            for (int c = 0; c < 4; ++c)                  // (k>>1)*1024
                b[c] = *(const v2f*)(mp + (size_t)(k >> 1) * (OUT_DIM * 2) + c * 32);
            #pragma unroll
            for (int c = 0; c < 4; ++c)
                acc[c] = wmma4(a, b[c], acc[c]);
        }

        #pragma unroll
        for (int c = 0; c < 4; ++c) {
            const int colg = (wave * 4 + c) * 16 + lq;
            #pragma unroll
            for (int j = 0; j < 8; ++j)
                dst[(j + 8 * half) * ZSTRIDE + colg] = acc[c][j];
        }
        __syncthreads();
    }

    // acc holds Z_100 tiles; emit pair-packed B layout:
    // zpack[((zr>>1)*512 + n)*2 + (zr&1)] = Z[zr][n]
    #pragma unroll
    for (int c = 0; c < 4; ++c) {
        const int colg = (wave * 4 + c) * 16 + lq;
        #pragma unroll
        for (int j = 0; j < 8; ++j) {
            const int zr = row0 + j + 8 * half;
            zpack[((((size_t)zr >> 1) << 9) + colg) * 2 + (zr & 1)] = acc[c][j];
        }
    }
}

// K3: y = x @ Z_100.  256 WGs x 64 rows; 16 acc tiles per wave; no LDS.
__global__ __launch_bounds__(256, 1)
void NSM_gemm(const float* __restrict__ x,
              const float* __restrict__ zpack,
              float* __restrict__ out)
{
    const int tid  = threadIdx.x;
    const int wave = tid >> 5;
    const int lane = tid & 31;
    const int half = lane >> 4;
    const int lq   = lane & 15;
    const int row0 = blockIdx.x * 64;

    v8f acc[4][4] = {};   // [row-tile][col-block]

    const float* xp = x + (size_t)(row0 + lq) * IN_DIM + 2 * half;
    const float* zp = zpack + ((size_t)half * OUT_DIM + wave * 64 + lq) * 2;

    #pragma unroll 2
    for (int k = 0; k < IN_DIM; k += 4) {
        v2f a[4], b[4];
        #pragma unroll
        for (int rt = 0; rt < 4; ++rt)
            a[rt] = *(const v2f*)(xp + rt * 16 * IN_DIM + k);
        #pragma unroll
        for (int c = 0; c < 4; ++c)
            b[c] = *(const v2f*)(zp + (size_t)(k >> 1) * (OUT_DIM * 2) + c * 32);
        #pragma unroll
        for (int rt = 0; rt < 4; ++rt)
            #pragma unroll
            for (int c = 0; c < 4; ++c)
                acc[rt][c] = wmma4(a[rt], b[c], acc[rt][c]);
    }

    #pragma unroll
    for (int rt = 0; rt < 4; ++rt)
        #pragma unroll
        for (int c = 0; c < 4; ++c) {
            const int colg = (wave * 4 + c) * 16 + lq;
            #pragma unroll
            for (int j = 0; j < 8; ++j)
                out[(size_t)(row0 + rt * 16 + j + 8 * half) * OUT_DIM + colg]
                    = acc[rt][c][j];
        }
}

extern "C" void kernel_launch(void* const* d_in, const int* in_sizes, int n_in,
                              void* d_out, int out_size, void* d_ws, size_t ws_size,
                              hipStream_t stream) {
    const float* x = (const float*)d_in[0];   // [16384, 1024]
    const float* W = (const float*)d_in[1];   // [512, 1024]
    const float* M = (const float*)d_in[2];   // [512, 512]
    float* out = (float*)d_out;               // [16384, 512]

    float* mneg  = (float*)d_ws;                                   // 1 MB
    float* zpack = (float*)d_ws + (size_t)OUT_DIM * OUT_DIM;       // 2 MB

    hipLaunchKernelGGL(NSM_packM, dim3((OUT_DIM * OUT_DIM) / 256), dim3(256),
                       0, stream, M, mneg);

    const size_t shmem = (size_t)2 * 16 * ZSTRIDE * sizeof(float); // 66048 B
    hipLaunchKernelGGL(NSM_zIter, dim3(IN_DIM / 16), dim3(256),
                       shmem, stream, W, mneg, zpack);

    hipLaunchKernelGGL(NSM_gemm, dim3(BATCH / 64), dim3(256),
                       0, stream, x, zpack, out);
}